// GCN_TME_34608846471581
// MI455X (gfx1250) — compile-verified
//
#include <hip/hip_runtime.h>

// Problem constants (from reference)
#define NN   100000
#define NE   1600000
#define CH   64
#define GWc  0.8f
#define EPSc 1e-5f

typedef __attribute__((ext_vector_type(2))) float v2f;
typedef __attribute__((ext_vector_type(8))) float v8f;

// ---------------- init: deg = 1 (self loop), zero stat accumulators ----------------
__global__ void k_init(float* __restrict__ acc, float* __restrict__ deg) {
    int i = blockIdx.x * blockDim.x + threadIdx.x;
    if (i < NN) deg[i] = 1.0f;           // self-loop contributes 1 to every degree
    if (i < 4)  acc[i] = 0.0f;           // sum / sumsq accumulators (re-zeroed every call)
}

// ---------------- degree: atomic count of incoming edges ----------------
__global__ void k_degree(const int* __restrict__ dst, float* __restrict__ deg) {
    int e = blockIdx.x * blockDim.x + threadIdx.x;
    if (e < NE) atomicAdd(&deg[dst[e]], 1.0f);
}

// ---------------- dis = rsqrt(deg) in place ----------------
__global__ void k_rsqrt(float* __restrict__ deg) {
    int i = blockIdx.x * blockDim.x + threadIdx.x;
    if (i < NN) {
        float d = deg[i];
        deg[i] = (d > 0.0f) ? rsqrtf(d) : 0.0f;
    }
}

// ---------------- Y = X @ W  (X: NN x 64, W: 64 x 64), f32 WMMA 16x16x4 ----------------
// One wave computes a 16-row x 64-col slab. NN = 6250 * 16 exactly (no ragged tiles).
__global__ __launch_bounds__(256) void k_gemm(const float* __restrict__ X,
                                              const float* __restrict__ W,
                                              float* __restrict__ Y) {
    const int wave   = threadIdx.x >> 5;
    const int lane   = threadIdx.x & 31;
    const int rowblk = blockIdx.x * 8 + wave;        // wave-uniform
    if (rowblk >= NN / 16) return;                   // uniform branch: EXEC stays all-1s
    const int r0 = rowblk * 16;
    const int lh = lane & 15;                        // 0..15
    const int hi = lane >> 4;                        // 0 or 1

    v8f acc0 = {}, acc1 = {}, acc2 = {}, acc3 = {};
#pragma unroll
    for (int k0 = 0; k0 < CH; k0 += 4) {
        // A fragment (16x4): lanes 0-15 hold K=k0,k0+1 of row r0+lh; lanes 16-31 hold K=k0+2,k0+3
        const float* ap = X + (size_t)(r0 + lh) * CH + (k0 + 2 * hi);
        v2f a; a.x = ap[0]; a.y = ap[1];
        // B fragments (4x16 each): lanes 0-15 rows K=k0,k0+1; lanes 16-31 rows K=k0+2,k0+3
        const float* bp = W + (size_t)(k0 + 2 * hi) * CH + lh;
        v2f b0; b0.x = bp[0];  b0.y = bp[CH];
        v2f b1; b1.x = bp[16]; b1.y = bp[16 + CH];
        v2f b2; b2.x = bp[32]; b2.y = bp[32 + CH];
        v2f b3; b3.x = bp[48]; b3.y = bp[48 + CH];
        acc0 = __builtin_amdgcn_wmma_f32_16x16x4_f32(false, a, false, b0, (short)0, acc0, false, false);
        acc1 = __builtin_amdgcn_wmma_f32_16x16x4_f32(false, a, false, b1, (short)0, acc1, false, false);
        acc2 = __builtin_amdgcn_wmma_f32_16x16x4_f32(false, a, false, b2, (short)0, acc2, false, false);
        acc3 = __builtin_amdgcn_wmma_f32_16x16x4_f32(false, a, false, b3, (short)0, acc3, false, false);
    }
    // C/D layout: VGPR v -> row r0 + v + 8*hi, col = lh + 16*ntile
    float* yp = Y + (size_t)(r0 + 8 * hi) * CH + lh;
#pragma unroll
    for (int v = 0; v < 8; ++v) {
        yp[(size_t)v * CH +  0] = acc0[v];
        yp[(size_t)v * CH + 16] = acc1[v];
        yp[(size_t)v * CH + 32] = acc2[v];
        yp[(size_t)v * CH + 48] = acc3[v];
    }
}

// ---------------- out[i] = xw[i] * dis[i]^2  (self-loop term, non-atomic init) ----------------
__global__ void k_selfloop(const float* __restrict__ xw, const float* __restrict__ dis,
                           float* __restrict__ out) {
    int t = blockIdx.x * blockDim.x + threadIdx.x;   // NN*16 threads, 4 channels each
    if (t >= NN * 16) return;
    int node = t >> 4;
    int c0   = (t & 15) << 2;
    float en = dis[node] * dis[node];
    float4 v = *(const float4*)(xw + (size_t)node * CH + c0);
    float4 r; r.x = v.x * en; r.y = v.y * en; r.z = v.z * en; r.w = v.w * en;
    *(float4*)(out + (size_t)node * CH + c0) = r;
}

// ---------------- edge scatter: out[dst] += xw[src] * dis[src]*dis[dst] ----------------
__global__ void k_edgeagg(const int* __restrict__ src, const int* __restrict__ dst,
                          const float* __restrict__ dis, const float* __restrict__ xw,
                          float* __restrict__ out) {
    long long t = (long long)blockIdx.x * blockDim.x + threadIdx.x;
    if (t >= (long long)NE * 16) return;
    int e  = (int)(t >> 4);
    int c0 = ((int)t & 15) << 2;
    int s = src[e], d = dst[e];
    float en = dis[s] * dis[d];
    float4 v = *(const float4*)(xw + (size_t)s * CH + c0);
    float* o = out + (size_t)d * CH + c0;
    atomicAdd(o + 0, v.x * en);
    atomicAdd(o + 1, v.y * en);
    atomicAdd(o + 2, v.z * en);
    atomicAdd(o + 3, v.w * en);
}

// ---------------- h = relu(in + b); x = GW*h + (1-GW)*h ----------------
__global__ void k_relu_mix(const float* __restrict__ in, const float* __restrict__ b,
                           float* __restrict__ out) {
    int t = blockIdx.x * blockDim.x + threadIdx.x;
    if (t >= NN * 16) return;
    int node = t >> 4;
    int c0   = (t & 15) << 2;
    float4 v  = *(const float4*)(in + (size_t)node * CH + c0);
    float4 bb = *(const float4*)(b + c0);
    float4 h;
    h.x = fmaxf(v.x + bb.x, 0.0f); h.y = fmaxf(v.y + bb.y, 0.0f);
    h.z = fmaxf(v.z + bb.z, 0.0f); h.w = fmaxf(v.w + bb.w, 0.0f);
    h.x = GWc * h.x + (1.0f - GWc) * h.x; h.y = GWc * h.y + (1.0f - GWc) * h.y;
    h.z = GWc * h.z + (1.0f - GWc) * h.z; h.w = GWc * h.w + (1.0f - GWc) * h.w;
    *(float4*)(out + (size_t)node * CH + c0) = h;
}

// ---------------- x = relu(x + b) in place, accumulate sum & sumsq for LayerNorm ----------------
__global__ __launch_bounds__(256) void k_relu_reduce(const float* __restrict__ b,
                                                     float* __restrict__ x,
                                                     float* __restrict__ acc) {
    __shared__ float ssum[256];
    __shared__ float ssq[256];
    int t = blockIdx.x * 256 + threadIdx.x;
    float lsum = 0.0f, lsq = 0.0f;
    if (t < NN * 16) {
        int node = t >> 4;
        int c0   = (t & 15) << 2;
        float4 v  = *(float4*)(x + (size_t)node * CH + c0);
        float4 bb = *(const float4*)(b + c0);
        v.x = fmaxf(v.x + bb.x, 0.0f); v.y = fmaxf(v.y + bb.y, 0.0f);
        v.z = fmaxf(v.z + bb.z, 0.0f); v.w = fmaxf(v.w + bb.w, 0.0f);
        *(float4*)(x + (size_t)node * CH + c0) = v;
        lsum = v.x + v.y + v.z + v.w;
        lsq  = v.x * v.x + v.y * v.y + v.z * v.z + v.w * v.w;
    }
    ssum[threadIdx.x] = lsum;
    ssq[threadIdx.x]  = lsq;
    __syncthreads();
#pragma unroll
    for (int s = 128; s > 0; s >>= 1) {
        if (threadIdx.x < s) {
            ssum[threadIdx.x] += ssum[threadIdx.x + s];
            ssq[threadIdx.x]  += ssq[threadIdx.x + s];
        }
        __syncthreads();
    }
    if (threadIdx.x == 0) {
        atomicAdd(&acc[0], ssum[0]);
        atomicAdd(&acc[1], ssq[0]);
    }
}

// ---------------- final graph-wide LayerNorm ----------------
__global__ void k_norm(const float* __restrict__ acc, const float* __restrict__ lnw,
                       const float* __restrict__ lnb, float* __restrict__ x) {
    int t = blockIdx.x * blockDim.x + threadIdx.x;
    if (t >= NN * 16) return;
    const float cnt  = (float)NN * (float)CH;
    float mean = acc[0] / cnt;
    float var  = acc[1] / cnt - mean * mean;
    float inv  = rsqrtf(var + EPSc) * lnw[0];
    float bb   = lnb[0];
    int node = t >> 4;
    int c0   = (t & 15) << 2;
    float4 v = *(float4*)(x + (size_t)node * CH + c0);
    v.x = (v.x - mean) * inv + bb; v.y = (v.y - mean) * inv + bb;
    v.z = (v.z - mean) * inv + bb; v.w = (v.w - mean) * inv + bb;
    *(float4*)(x + (size_t)node * CH + c0) = v;
}

extern "C" void kernel_launch(void* const* d_in, const int* in_sizes, int n_in,
                              void* d_out, int out_size, void* d_ws, size_t ws_size,
                              hipStream_t stream) {
    const float* X   = (const float*)d_in[0];
    const int*   edg = (const int*)d_in[1];      // harness delivers integer inputs as int32
    const int*   src = edg;                      // edges[0,:]
    const int*   dst = edg + NE;                 // edges[1,:]
    const float* W1  = (const float*)d_in[2];
    const float* b1  = (const float*)d_in[3];
    const float* W2  = (const float*)d_in[4];
    const float* b2  = (const float*)d_in[5];
    const float* lnw = (const float*)d_in[6];
    const float* lnb = (const float*)d_in[7];
    float* out = (float*)d_out;

    // workspace layout: [acc(4) | dis(NN) | buf1(NN*CH)]  (~26 MB)
    float* wsf  = (float*)d_ws;
    float* acc  = wsf;
    float* dis  = wsf + 4;
    float* buf1 = wsf + 4 + NN;

    const int nthr_n  = NN * 16;                       // node x 4-channel-groups
    const int gN      = (NN + 255) / 256;
    const int gE      = (NE + 255) / 256;
    const int gNC     = (nthr_n + 255) / 256;
    const int gEC     = (int)(((long long)NE * 16 + 255) / 256);
    const int gGemm   = (NN / 16 + 7) / 8;             // 8 waves per block

    // normalization setup
    k_init  <<<gN, 256, 0, stream>>>(acc, dis);
    k_degree<<<gE, 256, 0, stream>>>(dst, dis);
    k_rsqrt <<<gN, 256, 0, stream>>>(dis);

    // ----- layer 1: h = relu(Agg(X@W1) + b1) -----
    k_gemm    <<<gGemm, 256, 0, stream>>>(X, W1, out);            // xw1 -> out
    k_selfloop<<<gNC,   256, 0, stream>>>(out, dis, buf1);        // buf1 = self-loop term
    k_edgeagg <<<gEC,   256, 0, stream>>>(src, dst, dis, out, buf1);
    k_relu_mix<<<gNC,   256, 0, stream>>>(buf1, b1, out);         // h -> out

    // ----- layer 2: x = relu(Agg(h@W2) + b2) -----
    k_gemm    <<<gGemm, 256, 0, stream>>>(out, W2, buf1);         // xw2 -> buf1
    k_selfloop<<<gNC,   256, 0, stream>>>(buf1, dis, out);
    k_edgeagg <<<gEC,   256, 0, stream>>>(src, dst, dis, buf1, out);
    k_relu_reduce<<<gNC, 256, 0, stream>>>(b2, out, acc);         // in-place relu + stats

    // ----- graph-wide LayerNorm -----
    k_norm<<<gNC, 256, 0, stream>>>(acc, lnw, lnb, out);
}